// GCNEncoder_9646496547160
// MI455X (gfx1250) — compile-verified
//
#include <hip/hip_runtime.h>

typedef float v2f __attribute__((ext_vector_type(2)));
typedef float v4f __attribute__((ext_vector_type(4)));
typedef float v8f __attribute__((ext_vector_type(8)));

#define DDIM 256
#define LDS_STRIDE (DDIM + 4)   // 260 floats: 16B-aligned rows, bank stride 4 -> conflict-free

// ---------------------------------------------------------------------------
// Stage 1: h[n][j] = sum_k x[n][k] * W[j][k] + b[j]   via V_WMMA_F32_16X16X4_F32
// One wave per 16x16 tile of h. 8 waves/block share one M-tile (A staged in LDS).
// ---------------------------------------------------------------------------
__global__ __launch_bounds__(256) void gcn_gemm_wmma(
    const float* __restrict__ x, const float* __restrict__ W,
    const float* __restrict__ bias, float* __restrict__ h)
{
    __shared__ float As[16 * LDS_STRIDE];

    const int wave = threadIdx.x >> 5;
    const int lane = threadIdx.x & 31;

    const int tile0   = blockIdx.x * 8;        // 8 tiles per block, same mTile
    const int mTile   = tile0 >> 4;
    const int nTile   = (tile0 & 15) + wave;
    const int rowBase = mTile * 16;
    const int colBase = nTile * 16;

    // Cooperative coalesced load of A tile: 16 rows x 256 cols (16 KB)
    {
        const int r  = threadIdx.x >> 4;         // 0..15
        const int c0 = (threadIdx.x & 15) << 4;  // 16 floats per thread
        const v4f* s4 = (const v4f*)(x + (size_t)(rowBase + r) * DDIM + c0);
        v4f* d4 = (v4f*)(&As[r * LDS_STRIDE + c0]);
        d4[0] = s4[0]; d4[1] = s4[1]; d4[2] = s4[2]; d4[3] = s4[3];
    }
    __syncthreads();

    const int hi = lane >> 4;     // half-wave select (K pair)
    const int lr = lane & 15;

    // A fragment source (LDS): row lr, k = k0 + 2*hi  (float2)
    const float* Abase = &As[lr * LDS_STRIDE + hi * 2];
    // B fragment source (global W): B[k][j] = W[j][k] -> W row (colBase+lr), k contiguous
    const float* Bbase = W + (size_t)(colBase + lr) * DDIM + hi * 2;

    v8f acc = {};
#pragma unroll
    for (int k0 = 0; k0 < DDIM; k0 += 4) {
        v2f a = *(const v2f*)(Abase + k0);
        v2f bb = *(const v2f*)(Bbase + k0);
        // D = A(16x4) x B(4x16) + C : v_wmma_f32_16x16x4_f32
        acc = __builtin_amdgcn_wmma_f32_16x16x4_f32(
            /*neg_a=*/false, a, /*neg_b=*/false, bb,
            /*c_mod=*/(short)0, acc, /*reuse_a=*/false, /*reuse_b=*/false);
    }

    // Epilogue: D layout -> lane lr holds col (colBase+lr), rows v + 8*hi
    const float bv = bias[colBase + lr];
    float* hp = h + (size_t)rowBase * DDIM + colBase + lr;
#pragma unroll
    for (int v = 0; v < 8; ++v)
        hp[(size_t)(v + 8 * hi) * DDIM] = acc[v] + bv;
}

// ---------------------------------------------------------------------------
// Stage 0: zero the accumulator buffer (atomics require clean state each call)
// ---------------------------------------------------------------------------
__global__ void gcn_zero(float4* __restrict__ out, int n4)
{
    int i = blockIdx.x * blockDim.x + threadIdx.x;
    if (i < n4) out[i] = make_float4(0.f, 0.f, 0.f, 0.f);
}

// ---------------------------------------------------------------------------
// Stage 2: SpMM scatter-add. One wave32 per edge; 8 stride-32 floats per lane.
// h and out both fit in the 192MB L2, so gathers + atomics stay on-chip.
// ---------------------------------------------------------------------------
__global__ __launch_bounds__(256) void gcn_spmm(
    const int* __restrict__ rows, const int* __restrict__ cols,
    const float* __restrict__ vals, const float* __restrict__ h,
    float* __restrict__ out, int E)
{
    const int gw   = (blockIdx.x * 256 + threadIdx.x) >> 5;   // edge id
    const int lane = threadIdx.x & 31;
    if (gw >= E) return;

    const int   r = rows[gw];
    const int   c = cols[gw];
    const float v = vals[gw];

    const float* hp = h   + (size_t)c * DDIM;
    float*       op = out + (size_t)r * DDIM;

#pragma unroll
    for (int i = 0; i < 8; ++i) {
        const int j = (i << 5) + lane;        // coalesced 128B per step
        atomicAdd(op + j, v * hp[j]);         // global_atomic_add_f32
    }
}

// ---------------------------------------------------------------------------
// Stage 3: ReLU
// ---------------------------------------------------------------------------
__global__ void gcn_relu(float4* __restrict__ out, int n4)
{
    int i = blockIdx.x * blockDim.x + threadIdx.x;
    if (i < n4) {
        float4 v = out[i];
        v.x = fmaxf(v.x, 0.f); v.y = fmaxf(v.y, 0.f);
        v.z = fmaxf(v.z, 0.f); v.w = fmaxf(v.w, 0.f);
        out[i] = v;
    }
}

extern "C" void kernel_launch(void* const* d_in, const int* in_sizes, int n_in,
                              void* d_out, int out_size, void* d_ws, size_t ws_size,
                              hipStream_t stream)
{
    const float* x        = (const float*)d_in[0];
    const int*   adj_row  = (const int*)  d_in[1];
    const int*   adj_col  = (const int*)  d_in[2];
    const float* adj_vals = (const float*)d_in[3];
    const float* W        = (const float*)d_in[4];
    const float* b        = (const float*)d_in[5];
    float*       out      = (float*)d_out;
    float*       h        = (float*)d_ws;         // [N,256] fp32 = 102.4 MB

    const int N = in_sizes[0] / DDIM;             // 100000
    const int E = in_sizes[1];                    // 3200000

    const int n  = N * DDIM;
    const int n4 = n / 4;

    // 0) zero accumulator
    gcn_zero<<<(n4 + 255) / 256, 256, 0, stream>>>((float4*)out, n4);

    // 1) h = x @ W^T + b   (N/16 mTiles * 16 nTiles tiles, 8 waves per block)
    const int tiles  = (N / 16) * 16;             // == N here (N % 16 == 0)
    gcn_gemm_wmma<<<tiles / 8, 256, 0, stream>>>(x, W, b, h);

    // 2) scatter-add over edges (one wave per edge, 8 edges per block)
    gcn_spmm<<<(E + 7) / 8, 256, 0, stream>>>(adj_row, adj_col, adj_vals, h, out, E);

    // 3) ReLU
    gcn_relu<<<(n4 + 255) / 256, 256, 0, stream>>>((float4*)out, n4);
}